// ConvCaps_20761871909455
// MI455X (gfx1250) — compile-verified
//
#include <hip/hip_runtime.h>
#include <hip/hip_bf16.h>
#include <math.h>

typedef __attribute__((ext_vector_type(2))) float v2f;
typedef __attribute__((ext_vector_type(8))) float v8f;

#define EPSF 1e-10f
#define HALF_LN_2PI 0.9189385332046727f

// V is stored with the hh axis PERMUTED: slot h_s = h2*4 + h1 (instead of
// h1*4 + h2). All EM reductions sum over the whole hh axis, so they are
// permutation-invariant; mu/sig intermediates use the same permuted slots.
// Only the final mu written to d_out is de-permuted.

// ---------------------------------------------------------------------------
// Kernel 1: votes via V_WMMA_F32_16X16X4_F32.
// For each ikk = (Bi,ki,kj): bigA (128x4) = stacked W[Bi,ki,kj,Ci] 4x4 blocks,
// bigB (4x1152) = concatenated win 4x4 blocks over (b,wi',wj').
// D = bigA @ bigB holds ALL (Ci, pos) pair products = votes.
// With the permuted V layout each lane stores its 8 accumulators as TWO
// contiguous float4 (global_store_b128), fully coalesced across the wave.
// Grid: 288 blocks x 256 threads (8 waves); wave w owns m-tile w, loops n-tiles.
// ---------------------------------------------------------------------------
__global__ void votes_wmma_kernel(const float* __restrict__ poses,
                                  const float* __restrict__ W,
                                  float* __restrict__ V)
{
    const int ikk  = blockIdx.x;            // (Bi*3+ki)*3+kj
    const int Bi   = ikk / 9;
    const int ki   = (ikk / 3) % 3;
    const int kj   = ikk % 3;
    const int lane = threadIdx.x & 31;
    const int wv   = threadIdx.x >> 5;      // 0..7 -> m-tile
    const int mbase = wv * 16;

    // A tile (16x4 f32): lanes 0-15 rows m, K={0,1}; lanes 16-31 rows m, K={2,3}
    const int mrow = mbase + (lane & 15);   // m = Ci*4 + h1
    const int k0   = (lane < 16) ? 0 : 2;
    const float* Wp = W + (size_t)ikk * 512 + (size_t)mrow * 4 + k0;
    v2f A;
    A.x = Wp[0];
    A.y = Wp[1];

    // Per-lane D-store geometry (independent of nt except the pos offset):
    const int nlo    = lane & 15;
    const int mh     = (lane < 16) ? 0 : 8;
    const int CiBase = (mbase + mh) >> 2;   // C row of register group 0

    for (int nt = 0; nt < 72; ++nt) {
        // B tile (4x16 f32): col n = nt*16 + (lane&15), rows K=k0,k0+1
        const int n   = nt * 16 + (lane & 15);
        const int pos = n >> 2;             // b*36 + wi'*6 + wj'
        const int h2  = n & 3;
        const int bb  = pos / 36;
        const int rem = pos % 36;
        const int wip = rem / 6;
        const int wjp = rem % 6;
        v2f Bt;
        #pragma unroll
        for (int t = 0; t < 2; ++t) {
            const int k  = k0 + t;
            // reference's flat reinterpretation: f = wi'*96 + wj'*16 + k*4 + h2
            const int f  = wip * 96 + wjp * 16 + k * 4 + h2;
            const int hh = f / 36;
            const int fr = f % 36;
            const int wi = fr / 6;
            const int wj = fr % 6;
            const float v = poses[(((size_t)bb * 16 + hh) * 32 + Bi) * 196
                                  + (size_t)(2 * wi + ki) * 14 + (2 * wj + kj)];
            if (t == 0) Bt.x = v; else Bt.y = v;
        }

        v8f C = {};
        C = __builtin_amdgcn_wmma_f32_16x16x4_f32(
                /*neg_a=*/false, A, /*neg_b=*/false, Bt,
                /*c_mod=*/(short)0, C, /*reuse_a=*/false, /*reuse_b=*/false);

        // Store D tile: lane owns (ng -> pos2, hb); register r = group*4 + h1.
        // Permuted slot = hb*4 + h1 -> each group of 4 regs is contiguous.
        const int ng = nt * 16 + nlo;       // pos*4 + h2
        const int p2 = ng >> 2, hb = ng & 3;
        const int b2 = p2 / 36, r2 = p2 % 36;
        #pragma unroll
        for (int g = 0; g < 2; ++g) {
            const int j = (CiBase + g) * 36 + r2;
            float4 val;
            val.x = C[g * 4 + 0];
            val.y = C[g * 4 + 1];
            val.z = C[g * 4 + 2];
            val.w = C[g * 4 + 3];
            *reinterpret_cast<float4*>(
                &V[(((size_t)b2 * 288 + ikk) * 1152 + j) * 16 + hb * 4]) = val;
        }
    }
}

// ---------------------------------------------------------------------------
// Kernel 2: activation window gather. AW[b][i][pos], 82944 elements.
// ---------------------------------------------------------------------------
__global__ void aw_kernel(const float* __restrict__ act, float* __restrict__ AW)
{
    const int g = blockIdx.x * blockDim.x + threadIdx.x;
    if (g >= 8 * 288 * 36) return;
    const int pos = g % 36;
    const int t   = g / 36;
    const int i   = t % 288;
    const int b   = t / 288;
    const int Bi = i / 9, ki = (i / 3) % 3, kj = i % 3;
    const int wi = pos / 6, wj = pos % 6;
    AW[g] = act[((size_t)b * 32 + Bi) * 196 + (2 * wi + ki) * 14 + (2 * wj + kj)];
}

// ---------------------------------------------------------------------------
// Kernel 3: M-step. One thread per (b,j,h_s); 16 consecutive threads share
// (b,j) so the hh-reduction for cost is a 4-step wave32 shuffle. Single pass
// over i using E[v^2]-mu^2 for sigma. Grid: 576 x 256.
// permute_mu=1 (final iteration): de-permute the hh slot when writing mu.
// ---------------------------------------------------------------------------
__global__ void mstep_kernel(const float* __restrict__ V,
                             const float* __restrict__ AW,
                             const float* __restrict__ R,
                             const float* __restrict__ lambda_p,
                             const float* __restrict__ beta_v,
                             const float* __restrict__ beta_a,
                             float* __restrict__ mu,
                             float* __restrict__ sig,
                             float* __restrict__ a_out,
                             int first, int permute_mu)
{
    const int g  = blockIdx.x * blockDim.x + threadIdx.x;   // 147456
    const int h  = g & 15;            // permuted slot h_s
    const int bj = g >> 4;            // 0..9215
    const int j  = bj % 1152;
    const int b  = bj / 1152;
    const int pos = j % 36;
    const int Ci  = j / 36;

    const size_t base = (size_t)b * 288 * 1152;
    float s0 = 0.f, s1 = 0.f, s2 = 0.f;
    for (int i = 0; i < 288; ++i) {
        const size_t off = base + (size_t)i * 1152 + j;
        if (i + 1 < 288)
            __builtin_prefetch(&V[(off + 1152) * 16 + h], 0, 0);
        float rw = first ? (1.0f / 32.0f) : R[off];
        rw *= AW[((size_t)b * 288 + i) * 36 + pos];
        const float v = V[off * 16 + h];
        s0 += rw;
        s1 += rw * v;
        s2 += rw * v * v;
    }
    const float m  = s1 / s0;
    float ss = s2 / s0 - m * m;
    ss = fmaxf(ss, 0.0f);

    const int hout = permute_mu ? (((h & 3) << 2) | (h >> 2)) : h;
    mu[(size_t)bj * 16 + hout] = m;
    sig[(size_t)bj * 16 + h]   = ss;

    float costh = (beta_v[Ci] + logf(sqrtf(ss) + EPSF)) * s0;
    #pragma unroll
    for (int msk = 1; msk < 16; msk <<= 1)
        costh += __shfl_xor(costh, msk, 32);

    if (h == 0) {
        const float lam = lambda_p[0];
        const float x   = lam * (beta_a[Ci] - costh);
        a_out[bj] = 1.0f / (1.0f + __expf(-x));
    }
}

// ---------------------------------------------------------------------------
// Kernel 4: E-step. One block per (b,i); 128 threads x 9 j's = 1152 exactly.
// ap kept in registers; block-wide sum via wave shuffle + LDS.
// mu/sig/V all use the same permuted hh slots -> consistent.
// ---------------------------------------------------------------------------
__global__ void estep_kernel(const float* __restrict__ V,
                             const float* __restrict__ mu,
                             const float* __restrict__ sig,
                             const float* __restrict__ a_in,
                             float* __restrict__ R)
{
    const int bi  = blockIdx.x;        // b*288 + i, 2304 blocks
    const int b   = bi / 288;
    const int tid = threadIdx.x;       // 128

    float ap[9];
    float local = 0.f;
    #pragma unroll
    for (int t = 0; t < 9; ++t) {
        const int j = tid + t * 128;
        const size_t mb = ((size_t)b * 1152 + j) * 16;
        const size_t vb = ((size_t)bi * 1152 + j) * 16;
        float ps = 0.f;
        #pragma unroll
        for (int h = 0; h < 16; ++h) {
            const float v  = V[vb + h];
            const float m  = mu[mb + h];
            const float s  = sig[mb + h] + EPSF;
            const float d  = v - m;
            const float lp = -(d * d) / (2.0f * s) - 0.5f * logf(s) - HALF_LN_2PI;
            ps += __expf(lp);
        }
        const float apv = a_in[(size_t)b * 1152 + j] * ps;
        ap[t]  = apv;
        local += apv;
    }

    __shared__ float red[4];
    #pragma unroll
    for (int msk = 1; msk < 32; msk <<= 1)
        local += __shfl_xor(local, msk, 32);
    if ((tid & 31) == 0) red[tid >> 5] = local;
    __syncthreads();
    const float total = red[0] + red[1] + red[2] + red[3];
    const float inv   = 1.0f / (total + EPSF);

    #pragma unroll
    for (int t = 0; t < 9; ++t) {
        const int j = tid + t * 128;
        R[(size_t)bi * 1152 + j] = ap[t] * inv + EPSF;
    }
}

// ---------------------------------------------------------------------------
extern "C" void kernel_launch(void* const* d_in, const int* in_sizes, int n_in,
                              void* d_out, int out_size, void* d_ws, size_t ws_size,
                              hipStream_t stream)
{
    const float* poses    = (const float*)d_in[0];
    const float* act      = (const float*)d_in[1];
    const float* lambda_p = (const float*)d_in[2];
    const float* W        = (const float*)d_in[3];
    const float* beta_v   = (const float*)d_in[4];
    const float* beta_a   = (const float*)d_in[5];

    float* ws = (float*)d_ws;
    float* V  = ws;                  size_t o = 42467328;   // 8*288*1152*16
    float* AW = ws + o;              o += 82944;            // 8*288*36
    float* R  = ws + o;              o += 2654208;          // 8*288*1152
    float* mu = ws + o;              o += 147456;           // 8*1152*16
    float* sg = ws + o;              o += 147456;
    float* aa = ws + o;              o += 9216;             // 8*1152

    float* mu_out = (float*)d_out;            // (8,32,6,6,16) = 147456 floats
    float* a_out  = (float*)d_out + 147456;   // (8,32,6,6)    = 9216 floats

    votes_wmma_kernel<<<288, 256, 0, stream>>>(poses, W, V);
    aw_kernel<<<(8 * 288 * 36 + 255) / 256, 256, 0, stream>>>(act, AW);

    // iteration 0 (R = 1/C implicit)
    mstep_kernel<<<576, 256, 0, stream>>>(V, AW, R, lambda_p, beta_v, beta_a,
                                          mu, sg, aa, 1, 0);
    estep_kernel<<<2304, 128, 0, stream>>>(V, mu, sg, aa, R);

    // iteration 1
    mstep_kernel<<<576, 256, 0, stream>>>(V, AW, R, lambda_p, beta_v, beta_a,
                                          mu, sg, aa, 0, 0);
    estep_kernel<<<2304, 128, 0, stream>>>(V, mu, sg, aa, R);

    // iteration 2: write mu (de-permuted) and a straight into d_out
    mstep_kernel<<<576, 256, 0, stream>>>(V, AW, R, lambda_p, beta_v, beta_a,
                                          mu_out, sg, a_out, 0, 1);
}